// Steerable_CNP_81655918232149
// MI455X (gfx1250) — compile-verified
//
#include <hip/hip_runtime.h>

// CDNA5 (gfx1250) Steerable-CNP target head.
// K[t,g] generated in-register with v_exp_f32, consumed by
// V_WMMA_F32_16X16X4_F32. Both operand tables are PRE-SWIZZLED:
//  - Btab: B-fragment (value columns + ones column) in WMMA lane order,
//    one coalesced global_load_b64 per WMMA.
//  - Gtab: grid coords transposed {s*gx0, s*gx1, s*gy0, s*gy1} per chunk-half
//    with the length-scale factor s = sqrt(0.5*log2(e))/l baked in: the exp2
//    argument for both K-subrows is 4 packed VALU ops with free neg modifiers.
// The main loop is explicitly double-buffered (batches of 4 WMMAs, loads for
// batch i+1 issued before computing batch i) so no in-wave load stalls, and
// exps are computed a full batch of 4 ahead of their consuming WMMAs so the
// TRANS->WMMA hazard costs nothing.

typedef __attribute__((ext_vector_type(2))) float v2f;
typedef __attribute__((ext_vector_type(8))) float v8f;

#define G_PTS  16384
#define T_PTS  8192
#define TILES  (T_PTS / 16)     // 512 target tiles of 16
#define SPLIT  4                // grid-dimension split per tile
#define GCHUNK (G_PTS / SPLIT)  // 4096 grid points per wave
#define NCHUNK (G_PTS / 4)      // 4096 K-chunks of 4 grid points
#define CPB    4                // chunks (WMMAs) per pipelined batch

// arg = -((s*dx)^2 + (s*dy)^2), s^2 = 0.5*log2(e)/l^2
#define HALF_LOG2E 0.72134752044448170368f

// ---------------------------------------------------------------------------
// Prep 1: pre-swizzled B-fragment table.
// Btab layout: [chunk][lane][2] floats; chunk = g/4.
// Lane l holds B columns n = l&15, K-rows kb = (l<16 ? 0 : 2) and kb+1.
// Column values: n<2 -> mean_n, n<4 -> softplus(raw sigma), n==4 -> 1, else 0.
// means_grid[g=x*128+y][c] = FM[c][y][x].
// ---------------------------------------------------------------------------
__global__ void cnp_prep_b_kernel(const float* __restrict__ fm,
                                  float* __restrict__ Btab)
{
    int tid = blockIdx.x * 256 + threadIdx.x;   // 0 .. NCHUNK*64-1
    int c      = tid >> 6;          // chunk
    int within = tid & 63;
    int l      = within >> 1;       // lane 0..31
    int r      = within & 1;        // K sub-row 0/1
    int n      = l & 15;            // B column
    int kb     = (l < 16) ? 0 : 2;
    int g      = c * 4 + kb + r;    // grid point index

    int x = g >> 7, y = g & 127;
    int base = y * 128 + x;

    float val;
    if (n < 2) {
        val = fm[n * 16384 + base];
    } else if (n < 4) {
        float f = fm[n * 16384 + base];
        val = fmaxf(f, 0.0f) + log1pf(expf(-fabsf(f)));   // softplus
    } else if (n == 4) {
        val = 1.0f;                  // denominator (ones) column
    } else {
        val = 0.0f;
    }
    Btab[(size_t)c * 64 + l * 2 + r] = val;
}

// ---------------------------------------------------------------------------
// Prep 2: transposed, scale-baked grid table.
// Gtab layout: [chunk][half][4] = {s*gx(p), s*gx(p+1), s*gy(p), s*gy(p+1)}
// where p = chunk*4 + half*2. half = lane>>4 in the main kernel.
// ---------------------------------------------------------------------------
__global__ void cnp_prep_g_kernel(const float* __restrict__ gridp,
                                  const float* __restrict__ lsp,
                                  float* __restrict__ Gtab)
{
    int tid = blockIdx.x * 256 + threadIdx.x;   // 0 .. NCHUNK*8-1
    int c    = tid >> 3;
    int r    = tid & 7;
    int half = r >> 2;
    int j    = r & 3;                // 0,1 -> gx of p,p+1 ; 2,3 -> gy of p,p+1
    int p    = c * 4 + half * 2 + (j & 1);

    float l = lsp[0];
    float s = sqrtf(HALF_LOG2E) / l;

    float v = (j < 2) ? gridp[p * 2 + 0] : gridp[p * 2 + 1];
    Gtab[(size_t)c * 8 + half * 4 + j] = s * v;
}

// ---------------------------------------------------------------------------
// Pipelined helpers (fully unrolled; arrays stay in VGPRs)
// ---------------------------------------------------------------------------
__device__ __forceinline__ void load_batch(const float4* __restrict__ gt4,
                                           const float2* __restrict__ bt,
                                           int c,
                                           float4 (&g)[CPB], float2 (&b)[CPB])
{
#pragma unroll
    for (int j = 0; j < CPB; ++j) {
        g[j] = gt4[(size_t)(c + j) * 2];   // b128 (broadcast per half-wave)
        b[j] = bt[(size_t)(c + j) * 32];   // b64, lane-coalesced
    }
}

__device__ __forceinline__ void do_batch(const float4 (&g)[CPB],
                                         const float2 (&b)[CPB],
                                         v2f txsv, v2f tysv,
                                         v8f& acc0, v8f& acc1)
{
    v2f av[CPB];
#pragma unroll
    for (int j = 0; j < CPB; ++j) {        // pass 1: all exps first
        v2f gxs = {g[j].x, g[j].y};
        v2f gys = {g[j].z, g[j].w};
        v2f dxs = txsv - gxs;
        v2f dys = tysv - gys;
        v2f arg = -dxs * dxs - dys * dys;  // pk_mul + pk_fma with neg mods
        av[j].x = __builtin_amdgcn_exp2f(arg.x);
        av[j].y = __builtin_amdgcn_exp2f(arg.y);
    }
#pragma unroll
    for (int j = 0; j < CPB; ++j) {        // pass 2: WMMAs (alt accumulators)
        v2f bv; bv.x = b[j].x; bv.y = b[j].y;
        if (j & 1)
            acc1 = __builtin_amdgcn_wmma_f32_16x16x4_f32(
                false, av[j], false, bv, (short)0, acc1, false, false);
        else
            acc0 = __builtin_amdgcn_wmma_f32_16x16x4_f32(
                false, av[j], false, bv, (short)0, acc0, false, false);
    }
}

// ---------------------------------------------------------------------------
// Main: one wave = one 16-target tile x one grid chunk.
// Writes deterministic partial sums: partial[split][t][0..4]
// ---------------------------------------------------------------------------
__global__ void __launch_bounds__(256)
cnp_wmma_kernel(const float* __restrict__ xt,
                const float* __restrict__ lsp,
                const float* __restrict__ Gtab,
                const float* __restrict__ Btab,
                float* __restrict__ partial)
{
    const int lane  = threadIdx.x & 31;
    const int wave  = blockIdx.x * 8 + (threadIdx.x >> 5);
    const int tile  = wave >> 2;            // 0..511
    const int split = wave & (SPLIT - 1);   // 0..3

    const int n    = lane & 15;             // A-row (target) index
    const int half = lane >> 4;             // K-row half (0: K=0,1  1: K=2,3)

    // Per-lane target coordinates scaled by s (A-matrix row M = lane&15)
    const int   t = tile * 16 + n;
    const float l = lsp[0];
    const float s = sqrtf(HALF_LOG2E) / l;
    const float txs = xt[t * 2 + 0] * s;
    const float tys = xt[t * 2 + 1] * s;
    const v2f txsv = {txs, txs};
    const v2f tysv = {tys, tys};

    const float4* gt4 = (const float4*)Gtab + half;     // half-fixed base
    const float2* bt  = (const float2*)Btab + lane;     // lane-fixed base

    v8f acc0 = {};
    v8f acc1 = {};
    const int cstart = split * (GCHUNK / 4);            // 1024 chunks/wave
    const int cend   = cstart + (GCHUNK / 4);

    float4 gA[CPB], gB[CPB];
    float2 bA[CPB], bB[CPB];
    load_batch(gt4, bt, cstart, gA, bA);                // prime the pipe

    for (int c = cstart; c < cend; c += 2 * CPB) {      // 128 iterations
        // prefetch batch i+1, compute batch i
        load_batch(gt4, bt, c + CPB, gB, bB);
        do_batch(gA, bA, txsv, tysv, acc0, acc1);
        // prefetch batch i+2 (wrap to cstart on the tail: dead, in-bounds)
        int cp = c + 2 * CPB;
        if (cp >= cend) cp = cstart;
        load_batch(gt4, bt, cp, gA, bA);
        do_batch(gB, bB, txsv, tysv, acc0, acc1);
    }

    v8f d;
#pragma unroll
    for (int r = 0; r < 8; ++r) d[r] = acc0[r] + acc1[r];

    // D layout: VGPR r, lanes 0-15 -> row r, col lane; lanes 16-31 -> row r+8
    if (n < 5) {
        int mbase = half << 3;             // 0 or 8
        int trow  = tile * 16 + mbase;
        float* p = partial + ((size_t)split * T_PTS + trow) * 8 + n;
#pragma unroll
        for (int r = 0; r < 8; ++r) p[r * 8] = d[r];
    }
}

// ---------------------------------------------------------------------------
// Finalize: reduce the 4 split-partials, divide by denominator column.
// out[0:16384) = means (T,2), out[16384:32768) = sigmas (T,2)
// ---------------------------------------------------------------------------
__global__ void cnp_finalize_kernel(const float* __restrict__ partial,
                                    float* __restrict__ out)
{
    int tIdx = blockIdx.x * 256 + threadIdx.x;   // 0..8191
    float s0 = 0.f, s1 = 0.f, s2 = 0.f, s3 = 0.f, s4 = 0.f;
#pragma unroll
    for (int sp = 0; sp < SPLIT; ++sp) {
        const float* p = partial + ((size_t)sp * T_PTS + tIdx) * 8;
        s0 += p[0]; s1 += p[1]; s2 += p[2]; s3 += p[3]; s4 += p[4];
    }
    float inv = 1.0f / s4;
    out[tIdx * 2 + 0] = s0 * inv;
    out[tIdx * 2 + 1] = s1 * inv;
    out[2 * T_PTS + tIdx * 2 + 0] = s2 * inv;
    out[2 * T_PTS + tIdx * 2 + 1] = s3 * inv;
}

extern "C" void kernel_launch(void* const* d_in, const int* in_sizes, int n_in,
                              void* d_out, int out_size, void* d_ws, size_t ws_size,
                              hipStream_t stream) {
    const float* fm    = (const float*)d_in[0];  // (4,128,128)
    const float* gridp = (const float*)d_in[1];  // (16384,2)
    const float* xt    = (const float*)d_in[2];  // (8192,2)
    const float* ls    = (const float*)d_in[3];  // scalar
    float* out = (float*)d_out;

    // ws layout: Btab [NCHUNK*64] floats (1MB), Gtab [NCHUNK*8] floats
    // (128KB), partial [SPLIT*T_PTS*8] floats (1MB).
    float* Btab    = (float*)d_ws;
    float* Gtab    = Btab + (size_t)NCHUNK * 64;
    float* partial = Gtab + (size_t)NCHUNK * 8;

    cnp_prep_b_kernel<<<(NCHUNK * 64) / 256, 256, 0, stream>>>(fm, Btab);
    cnp_prep_g_kernel<<<(NCHUNK * 8) / 256, 256, 0, stream>>>(gridp, ls, Gtab);
    cnp_wmma_kernel<<<(TILES * SPLIT) / 8, 256, 0, stream>>>(xt, ls, Gtab, Btab, partial);
    cnp_finalize_kernel<<<T_PTS / 256, 256, 0, stream>>>(partial, out);
}